// TensorProductUniform3x1d_4621384810538
// MI455X (gfx1250) — compile-verified
//
#include <hip/hip_runtime.h>
#include <stdint.h>

#define U     128
#define NSEG  32
#define NPATH 512
#define BATCH 2048
#define SEGF  (NSEG * U)            // 4096 floats per operand per batch
#define SEGB  (SEGF * 4)            // 16384 bytes per tile

typedef uint32_t v4u __attribute__((ext_vector_type(4)));
typedef int      v4i __attribute__((ext_vector_type(4)));
typedef int      v8i __attribute__((ext_vector_type(8)));

// ---------------------------------------------------------------------------
// Plan builder: stable counting sort of paths by pout.
//   c_s[i]   : coeff of i-th sorted path
//   o0_s[i]  : BYTE offset (p0*U*4) of its x0 segment
//   o1_s[i]  : BYTE offset (p1*U*4) of its x1 segment
//   bstart[s]: bucket start for output segment s (bstart[NSEG] == NPATH)
// ---------------------------------------------------------------------------
__global__ __launch_bounds__(NPATH) void build_plan_kernel(
    const float* __restrict__ coeffs,
    const int*   __restrict__ p0,
    const int*   __restrict__ p1,
    const int*   __restrict__ pout,
    float* __restrict__ c_s,
    int*   __restrict__ o0_s,
    int*   __restrict__ o1_s,
    int*   __restrict__ bstart)
{
    __shared__ int counts[NSEG];
    const int k  = threadIdx.x;          // 0..511
    const int my = pout[k];

    if (k < NSEG) {
        int c = 0;
        for (int j = 0; j < NPATH; ++j) c += (pout[j] == k) ? 1 : 0;
        counts[k] = c;
    }
    __syncthreads();

    // stable rank of path k within its pout bucket (uniform trip count)
    int rank = 0;
    for (int j = 0; j < NPATH; ++j) {
        const int pj = pout[j];
        rank += ((pj == my) && (j < k)) ? 1 : 0;
    }
    int start = 0;
    for (int s = 0; s < NSEG; ++s) start += (s < my) ? counts[s] : 0;

    const int dst = start + rank;
    c_s[dst]  = coeffs[k];
    o0_s[dst] = p0[k] * (U * 4);
    o1_s[dst] = p1[k] * (U * 4);

    if (k <= NSEG) {
        int st = 0;
        for (int s = 0; s < NSEG; ++s) st += (s < k) ? counts[s] : 0;
        bstart[k] = st;
    }
}

// ---------------------------------------------------------------------------
// TDM descriptor helper: 1-row tile of SEGF fp32 elements, contiguous,
// global -> LDS.  D# layout per CDNA5 ISA ch.8 (§8.3/§8.4).
// ---------------------------------------------------------------------------
#if __has_builtin(__builtin_amdgcn_tensor_load_to_lds)
#define TPU_USE_TDM 1
__device__ __forceinline__ void tdm_load_tile(const float* gptr, uint32_t lds_byte_off)
{
    const uint64_t ga = (uint64_t)(uintptr_t)gptr;
    // Group 0: [1:0]=count=1, [63:32]=lds_addr, [120:64]=global_addr, [127:126]=type=2
    v4u g0;
    g0.x = 1u;                                   // count=1, no gather
    g0.y = lds_byte_off;                         // lds_addr (bytes)
    g0.z = (uint32_t)(ga & 0xFFFFFFFFu);         // global_addr[31:0]
    g0.w = (uint32_t)((ga >> 32) & 0x01FFFFFFu)  // global_addr[56:32]
         | (2u << 30);                           // type=2 ("image")
    // Group 1: wg_mask=0 | data_size=2 (4B) ; tensor_dim0=SEGF ; tensor_dim1=1 ;
    //          tile_dim0=SEGF ; tile_dim1=1 ; tile_dim2=0 ;
    //          tensor_dim0_stride=SEGF ; tensor_dim1_stride=SEGF
    v8i g1;
    g1[0] = (int)(2u << 16);                     // data_size=4B, all flags 0
    g1[1] = (int)((uint32_t)(SEGF & 0xFFFF) << 16);   // tensor_dim0[15:0] @ [63:48]
    g1[2] = (int)((uint32_t)(1) << 16);               // dim0 hi=0 | tensor_dim1[15:0]
    g1[3] = (int)((uint32_t)(SEGF & 0xFFFF) << 16);   // dim1 hi=0 | tile_dim0
    g1[4] = 1;                                        // tile_dim1=1 | tile_dim2=0
    g1[5] = SEGF;                                     // tensor_dim0_stride lo32
    g1[6] = (int)((uint32_t)(SEGF & 0xFFFF) << 16);   // stride0 hi=0 | stride1 lo16
    g1[7] = 0;                                        // stride1 hi32
    v4i g2; g2[0] = 1; g2[1] = 0; g2[2] = 0; g2[3] = 0;   // tensor_dim2=1, rest 0
    v4i g3; g3[0] = 0; g3[1] = 0; g3[2] = 0; g3[3] = 0;
#if defined(__clang_major__) && (__clang_major__ >= 23)
    v8i g4; g4[0]=0; g4[1]=0; g4[2]=0; g4[3]=0; g4[4]=0; g4[5]=0; g4[6]=0; g4[7]=0;
    __builtin_amdgcn_tensor_load_to_lds(g0, g1, g2, g3, g4, 0);
#else
    __builtin_amdgcn_tensor_load_to_lds(g0, g1, g2, g3, 0);
#endif
}
#else
#define TPU_USE_TDM 0
#endif

// ---------------------------------------------------------------------------
// Main kernel: one block per batch element, 64 threads (2 wave32).
// Stage x0[b], x1[b] (16 KB each) into LDS with the Tensor Data Mover
// (fallback: async global->LDS loads), then per-output-segment register
// accumulation over the sorted path buckets.
// ---------------------------------------------------------------------------
__global__ __launch_bounds__(64) void tpu_seg_kernel(
    const float* __restrict__ x0,
    const float* __restrict__ x1,
    const float* __restrict__ c_s,
    const int*   __restrict__ o0_s,
    const int*   __restrict__ o1_s,
    const int*   __restrict__ bstart,
    float* __restrict__ out)
{
    __shared__ float s0[SEGF];
    __shared__ float s1[SEGF];

    const int b = blockIdx.x;
    const int t = threadIdx.x;          // 0..63, owns float2 at u = 2t
    const float* g0 = x0 + (size_t)b * SEGF;
    const float* g1 = x1 + (size_t)b * SEGF;

    // Group-relative LDS byte offsets (flat LDS pointers carry the offset in
    // the low 32 bits; aperture bits live in [63:32]).
    const uint32_t l0 = (uint32_t)(uintptr_t)s0;
    const uint32_t l1 = (uint32_t)(uintptr_t)s1;

#if TPU_USE_TDM
    if (t < 32) {                       // wave 0 issues the two tile DMAs
        tdm_load_tile(g0, l0);
        tdm_load_tile(g1, l1);
        __builtin_amdgcn_s_wait_tensorcnt(0);
    }
#else
    for (int i = t * 4; i < SEGF; i += 64 * 4) {
        asm volatile("global_load_async_to_lds_b128 %0, %1, off"
                     :: "v"(l0 + (uint32_t)i * 4u), "v"(g0 + i) : "memory");
        asm volatile("global_load_async_to_lds_b128 %0, %1, off"
                     :: "v"(l1 + (uint32_t)i * 4u), "v"(g1 + i) : "memory");
    }
    asm volatile("s_wait_asynccnt 0x0" ::: "memory");
#endif
    __syncthreads();

    float* o = out + (size_t)b * SEGF;
    const char* s0b = (const char*)s0 + (uint32_t)(t * 8);  // per-lane base
    const char* s1b = (const char*)s1 + (uint32_t)(t * 8);

    for (int s = 0; s < NSEG; ++s) {
        const int beg = bstart[s];
        const int end = bstart[s + 1];
        float accx = 0.0f, accy = 0.0f;
        for (int i = beg; i < end; ++i) {
            const float c  = c_s[i];          // uniform -> s_load
            const int   a0 = o0_s[i];         // uniform byte offset
            const int   a1 = o1_s[i];
            const float2 va = *(const float2*)(s0b + a0);   // ds_load_b64
            const float2 vb = *(const float2*)(s1b + a1);   // ds_load_b64
            accx = fmaf(c * va.x, vb.x, accx);
            accy = fmaf(c * va.y, vb.y, accy);
        }
        float2 r; r.x = accx; r.y = accy;
        *(float2*)(o + s * U + t * 2) = r;    // covers all outputs (zeros too)
    }
}

// ---------------------------------------------------------------------------
// Inputs (setup_inputs order): x0, x1, coeffs, p0, p1, pout
// Output: float32 [BATCH, NSEG*U]
// ---------------------------------------------------------------------------
extern "C" void kernel_launch(void* const* d_in, const int* in_sizes, int n_in,
                              void* d_out, int out_size, void* d_ws, size_t ws_size,
                              hipStream_t stream)
{
    const float* x0     = (const float*)d_in[0];
    const float* x1     = (const float*)d_in[1];
    const float* coeffs = (const float*)d_in[2];
    const int*   p0     = (const int*)d_in[3];
    const int*   p1     = (const int*)d_in[4];
    const int*   pout   = (const int*)d_in[5];

    float* c_s    = (float*)d_ws;
    int*   o0_s   = (int*)((char*)d_ws + NPATH * 4);
    int*   o1_s   = (int*)((char*)d_ws + NPATH * 8);
    int*   bstart = (int*)((char*)d_ws + NPATH * 12);

    build_plan_kernel<<<1, NPATH, 0, stream>>>(coeffs, p0, p1, pout,
                                               c_s, o0_s, o1_s, bstart);
    tpu_seg_kernel<<<BATCH, 64, 0, stream>>>(x0, x1, c_s, o0_s, o1_s, bstart,
                                             (float*)d_out);
}